// Core_crop_function_11055245820322
// MI455X (gfx1250) — compile-verified
//
#include <hip/hip_runtime.h>
#include <stdint.h>

// Pure gather: out[b*P + p, c] = imgs[b, c, y, x] with (x,y) = batch_points[b, p, :].
// Memory-bound: ~268 MB of scattered cache-line traffic -> ~11.5 us at 23.3 TB/s.
// CDNA5 paths used: global_load_async_to_lds_b64 (ASYNCcnt, double-buffered),
// s_wait_asynccnt, TH=NT loads/stores, wave32 scalarization via readfirstlane.

typedef float v4f __attribute__((ext_vector_type(4)));

#define B_      8
#define C_      128
#define H_      512
#define W_      512
#define P_      2048
#define PLANE_  (H_ * W_)        // 262144 elements = 1 MB stride between channels
#define NROWS_  (B_ * P_)        // 16384 output rows
#define NGROUPS_ (NROWS_ / 8)    // 8 points (waves) per 256-thread block
#define TOTAL_OUT_ (NROWS_ * C_) // 2097152 floats of crop data

__global__ __launch_bounds__(256)
void crop_gather_kernel(const float* __restrict__ imgs,
                        const int*   __restrict__ pts,   // (B, P, 2) int32: [x, y]
                        float*       __restrict__ out,
                        int out_size)
{
    // Double-buffered LDS staging area for 8 points (x,y) per group.
    __shared__ __align__(16) int pts_sm[2][8][2];

    const int wv   = __builtin_amdgcn_readfirstlane((int)(threadIdx.x >> 5)); // wave id 0..7 (SGPR)
    const int lane = threadIdx.x & 31;
    const int stride = gridDim.x;

    // Prologue: async-stage group g (blockIdx.x) point coords into LDS buffer 0.
    int g = blockIdx.x;
    if (wv == 0 && lane < 8 && g < NGROUPS_) {
        uint32_t lds   = (uint32_t)(uintptr_t)&pts_sm[0][lane][0];
        uint64_t gaddr = (uint64_t)(uintptr_t)(pts + ((size_t)g * 8 + lane) * 2);
        asm volatile("global_load_async_to_lds_b64 %0, %1, off"
                     :: "v"(lds), "v"(gaddr) : "memory");
    }

    int it = 0;
    for (; g < NGROUPS_; g += stride, ++it) {
        const int buf = it & 1;

        // Wait for the async stage of THIS group, then make it visible block-wide.
        asm volatile("s_wait_asynccnt 0x0" ::: "memory");
        __syncthreads();

        // Kick off the async stage of the NEXT group into the other buffer
        // while we do the heavy scattered gather for this one.
        const int gn = g + stride;
        if (wv == 0 && lane < 8 && gn < NGROUPS_) {
            uint32_t lds   = (uint32_t)(uintptr_t)&pts_sm[buf ^ 1][lane][0];
            uint64_t gaddr = (uint64_t)(uintptr_t)(pts + ((size_t)gn * 8 + lane) * 2);
            asm volatile("global_load_async_to_lds_b64 %0, %1, off"
                         :: "v"(lds), "v"(gaddr) : "memory");
        }

        // This wave's point (uniform across the wave -> force into SGPRs).
        const int row = g * 8 + wv;                   // 0..16383, scalar
        int x = pts_sm[buf][wv][0];
        int y = pts_sm[buf][wv][1];
        x = __builtin_amdgcn_readfirstlane(x);
        y = __builtin_amdgcn_readfirstlane(y);

        const int b = row >> 11;                      // row / P_
        const float* base = imgs + (size_t)b * C_ * PLANE_
                                 + (size_t)y * W_ + (size_t)x;

        // Lane l gathers channels 4l..4l+3 (1 MB apart; within 24-bit ioffset of p0),
        // non-temporal: each 128B line yields only 4 useful bytes, don't keep it.
        const int c0 = lane * 4;
        const float* p0 = base + (size_t)c0 * PLANE_;
        float a0 = __builtin_nontemporal_load(p0);
        float a1 = __builtin_nontemporal_load(p0 + 1 * PLANE_);
        float a2 = __builtin_nontemporal_load(p0 + 2 * PLANE_);
        float a3 = __builtin_nontemporal_load(p0 + 3 * PLANE_);

        // Coalesced 512B-per-wave non-temporal store (write-once stream).
        v4f v = {a0, a1, a2, a3};
        __builtin_nontemporal_store(v, (v4f*)(out + (size_t)row * C_ + c0));
    }

    // Second tuple element of the reference: length = NROWS_ (appended as float).
    if (blockIdx.x == 0 && threadIdx.x == 0 && out_size > TOTAL_OUT_) {
        out[TOTAL_OUT_] = (float)NROWS_;
    }
}

extern "C" void kernel_launch(void* const* d_in, const int* in_sizes, int n_in,
                              void* d_out, int out_size, void* d_ws, size_t ws_size,
                              hipStream_t stream)
{
    const float* imgs = (const float*)d_in[0];
    const int*   pts  = (const int*)d_in[1];
    float*       out  = (float*)d_out;

    // 1024 blocks x 2 groups each = 2048 groups (16384 waves of gather work);
    // 2-deep grid-stride loop exercises the async double-buffer.
    dim3 grid(NGROUPS_ / 2);
    dim3 block(256);
    hipLaunchKernelGGL(crop_gather_kernel, grid, block, 0, stream,
                       imgs, pts, out, out_size);
}